// SuperpixelEbli_23545010717580
// MI455X (gfx1250) — compile-verified
//
#include <hip/hip_runtime.h>
#include <hip/hip_bf16.h>

#define B_GRAPHS 1024

typedef float v2f __attribute__((ext_vector_type(2)));
typedef float v4f __attribute__((ext_vector_type(4)));
typedef float v8f __attribute__((ext_vector_type(8)));

__device__ __forceinline__ float leaky(float x) { return x >= 0.f ? x : 0.01f * x; }

// ---------------------------------------------------------------------------
// SpMM, F=3: one thread per nonzero. y[r][0..2] += v * x[c][0..2]
// ---------------------------------------------------------------------------
__global__ void spmm_f3(const int* __restrict__ rows, const int* __restrict__ cols,
                        const float* __restrict__ vals, const float* __restrict__ x,
                        float* __restrict__ y, int nnz)
{
    int e = blockIdx.x * blockDim.x + threadIdx.x;
    if (e >= nnz) return;
    int r = rows[e], c = cols[e];
    float v = vals[e];
    atomicAdd(&y[r * 3 + 0], v * x[c * 3 + 0]);
    atomicAdd(&y[r * 3 + 1], v * x[c * 3 + 1]);
    atomicAdd(&y[r * 3 + 2], v * x[c * 3 + 2]);
}

// ---------------------------------------------------------------------------
// Linear 3 -> 32 + LeakyReLU, writes into column block 0 of the (n,96) buffer.
// One thread per (node, out-feature).
// ---------------------------------------------------------------------------
__global__ void lin3_leaky(const float* __restrict__ y3, const float* __restrict__ W, // (32,3)
                           const float* __restrict__ b, float* __restrict__ O, int n)
{
    int idx = blockIdx.x * blockDim.x + threadIdx.x;
    int i = idx >> 5, j = idx & 31;
    if (i >= n) return;
    float acc = b[j]
              + y3[i * 3 + 0] * W[j * 3 + 0]
              + y3[i * 3 + 1] * W[j * 3 + 1]
              + y3[i * 3 + 2] * W[j * 3 + 2];
    O[(size_t)i * 96 + j] = leaky(acc);
}

// ---------------------------------------------------------------------------
// SpMM, F=32: one lane per (edge, 4-feature chunk). 8 lanes per edge.
// One b128 gather + 4 scalar atomics per lane. src has row stride srcStride
// (in floats); src rows and y rows are 16B-aligned.
// ---------------------------------------------------------------------------
__global__ void spmm_f32(const int* __restrict__ rows, const int* __restrict__ cols,
                         const float* __restrict__ vals,
                         const float* __restrict__ src, int srcStride,
                         float* __restrict__ y, int nnz)
{
    unsigned idx = blockIdx.x * blockDim.x + threadIdx.x;
    unsigned e = idx >> 3, f = (idx & 7) * 4;
    if (e >= (unsigned)nnz) return;
    int r = rows[e], c = cols[e];
    float v = vals[e];
    v4f s = *(const v4f*)(src + (size_t)c * srcStride + f);
    float* yp = y + (size_t)r * 32 + f;
    atomicAdd(yp + 0, v * s.x);
    atomicAdd(yp + 1, v * s.y);
    atomicAdd(yp + 2, v * s.z);
    atomicAdd(yp + 3, v * s.w);
}

// ---------------------------------------------------------------------------
// (n,32) @ W(32,32)^T + bias, LeakyReLU, store into strided output (ldo=96).
// fp32 WMMA: V_WMMA_F32_16X16X4_F32. Each wave: 16 rows x 32 cols,
// two 16x16 accumulators, K=32 in 8 steps of 4.
//   A layout (16x4 f32): lane L -> M=L&15; v0/v1 hold K=(L>>4)*2, +1
//   B layout (4x16 f32): lane L -> N=L&15; v0/v1 hold K=(L>>4)*2, +1
//     B[k][j] = W[j][k] (W row-major, (out,in)) -> contiguous float2 at W[j*32+k]
//   D layout (16x16 f32): VGPR r -> row r + (L>>4)*8, col L&15
// OOB A rows are clamped (not predicated) so EXEC stays all-ones with no
// divergence; OOB rows compute garbage that is never stored.
// ---------------------------------------------------------------------------
__global__ void gemm32_wmma_leaky(const float* __restrict__ A,   // n x 32, stride 32
                                  const float* __restrict__ W,   // 32 x 32 row-major
                                  const float* __restrict__ bias,
                                  float* __restrict__ Out, int ldo, int n)
{
    int wave = (blockIdx.x * blockDim.x + threadIdx.x) >> 5;
    int lane = threadIdx.x & 31;
    int m0 = wave * 16;
    if (m0 >= n) return;                 // uniform per wave

    v8f acc0 = {};
    v8f acc1 = {};

    int kh   = (lane >> 4) * 2;
    int colb = lane & 15;
    int rowA = m0 + (lane & 15);
    if (rowA > n - 1) rowA = n - 1;      // clamp: keep loads in-bounds, no divergence
    const float* Arow = A + (size_t)rowA * 32 + kh;
    const float* W0   = W + colb * 32 + kh;
    const float* W1   = W + (colb + 16) * 32 + kh;

#pragma unroll
    for (int k0 = 0; k0 < 32; k0 += 4) {
        v2f a  = *(const v2f*)(Arow + k0);
        v2f b0 = *(const v2f*)(W0 + k0);
        v2f b1 = *(const v2f*)(W1 + k0);
        acc0 = __builtin_amdgcn_wmma_f32_16x16x4_f32(false, a, false, b0, (short)0, acc0, false, false);
        acc1 = __builtin_amdgcn_wmma_f32_16x16x4_f32(false, a, false, b1, (short)0, acc1, false, false);
    }

    int col  = lane & 15;
    int half = (lane >> 4) * 8;
    float bc0 = bias[col], bc1 = bias[col + 16];
#pragma unroll
    for (int r = 0; r < 8; ++r) {
        int row = m0 + half + r;
        if (row < n) {
            Out[(size_t)row * ldo + col]      = leaky(acc0[r] + bc0);
            Out[(size_t)row * ldo + col + 16] = leaky(acc1[r] + bc1);
        }
    }
}

// ---------------------------------------------------------------------------
// Per-node (96 -> 10) linear, then atomic segment-sum pooling.
// ---------------------------------------------------------------------------
__global__ void layer_pool(const float* __restrict__ O, const float* __restrict__ W, // (10,96)
                           const float* __restrict__ b, const int* __restrict__ batch,
                           float* __restrict__ psum, float* __restrict__ pcnt, int n)
{
    int i = blockIdx.x * blockDim.x + threadIdx.x;
    if (i >= n) return;
    const float* row = O + (size_t)i * 96;
    float f[10];
#pragma unroll
    for (int j = 0; j < 10; ++j) f[j] = b[j];
    for (int k4 = 0; k4 < 96; k4 += 4) {
        v4f xv = *(const v4f*)(row + k4);
#pragma unroll
        for (int j = 0; j < 10; ++j) {
            f[j] += xv.x * W[j * 96 + k4 + 0];
            f[j] += xv.y * W[j * 96 + k4 + 1];
            f[j] += xv.z * W[j * 96 + k4 + 2];
            f[j] += xv.w * W[j * 96 + k4 + 3];
        }
    }
    int g = batch[i];
    atomicAdd(&pcnt[g], 1.0f);
#pragma unroll
    for (int j = 0; j < 10; ++j) atomicAdd(&psum[g * 10 + j], f[j]);
}

// ---------------------------------------------------------------------------
// Mean, concat(30), (30 -> 10) linear, softmax. One thread per graph.
// ---------------------------------------------------------------------------
__global__ void final_combine(const float* __restrict__ psum, const float* __restrict__ pcnt,
                              const float* __restrict__ cW,  // (10,30)
                              const float* __restrict__ cb, float* __restrict__ out)
{
    int g = blockIdx.x * blockDim.x + threadIdx.x;
    if (g >= B_GRAPHS) return;
    float m[30];
#pragma unroll
    for (int t = 0; t < 3; ++t) {
        float c = pcnt[t * B_GRAPHS + g];
        c = c > 1.f ? c : 1.f;
        float inv = 1.f / c;
        for (int j = 0; j < 10; ++j)
            m[t * 10 + j] = psum[((size_t)t * B_GRAPHS + g) * 10 + j] * inv;
    }
    float z[10], zmax = -1e30f;
#pragma unroll
    for (int j = 0; j < 10; ++j) {
        float a = cb[j];
        for (int k = 0; k < 30; ++k) a += m[k] * cW[j * 30 + k];
        z[j] = a;
        zmax = a > zmax ? a : zmax;
    }
    float s = 0.f;
#pragma unroll
    for (int j = 0; j < 10; ++j) { z[j] = expf(z[j] - zmax); s += z[j]; }
    float inv = 1.f / s;
#pragma unroll
    for (int j = 0; j < 10; ++j) out[g * 10 + j] = z[j] * inv;
}

// ---------------------------------------------------------------------------
extern "C" void kernel_launch(void* const* d_in, const int* in_sizes, int n_in,
                              void* d_out, int out_size, void* d_ws, size_t ws_size,
                              hipStream_t stream)
{
    const float* X[3]   = { (const float*)d_in[0], (const float*)d_in[1], (const float*)d_in[2] };
    const int*   Lr[3]  = { (const int*)d_in[3],  (const int*)d_in[6],  (const int*)d_in[9]  };
    const int*   Lc[3]  = { (const int*)d_in[4],  (const int*)d_in[7],  (const int*)d_in[10] };
    const float* Lv[3]  = { (const float*)d_in[5], (const float*)d_in[8], (const float*)d_in[11] };
    const int*   bat[3] = { (const int*)d_in[12], (const int*)d_in[13], (const int*)d_in[14] };
    int n[3]   = { in_sizes[0] / 3, in_sizes[1] / 3, in_sizes[2] / 3 };
    int nnz[3] = { in_sizes[3], in_sizes[6], in_sizes[9] };

    // ---- params: 15.. in nested-dict order. The nine C{t}_{s} (W,b) pairs
    // come first under both insertion and alphabetical flattening; the
    // combined/layer group order is disambiguated by element count
    // (combined_W = 300, layer*_W = 960).
    int p = 15;
    const float *CW1[3], *Cb1[3], *CW2[3], *Cb2[3], *CW3[3], *Cb3[3];
    for (int t = 0; t < 3; ++t) {
        CW1[t] = (const float*)d_in[p++]; Cb1[t] = (const float*)d_in[p++];
        CW2[t] = (const float*)d_in[p++]; Cb2[t] = (const float*)d_in[p++];
        CW3[t] = (const float*)d_in[p++]; Cb3[t] = (const float*)d_in[p++];
    }
    const float *layW[3], *layb[3], *combW, *combb;
    if (in_sizes[p] == 300) {            // alphabetical: combined_* first
        combW = (const float*)d_in[p++]; combb = (const float*)d_in[p++];
        for (int t = 0; t < 3; ++t) {
            layW[t] = (const float*)d_in[p++]; layb[t] = (const float*)d_in[p++];
        }
    } else {                             // insertion order: layer0..2 then combined
        for (int t = 0; t < 3; ++t) {
            layW[t] = (const float*)d_in[p++]; layb[t] = (const float*)d_in[p++];
        }
        combW = (const float*)d_in[p++]; combb = (const float*)d_in[p++];
    }

    // ---- workspace layout (towers processed sequentially, buffers reused)
    size_t nmax = (size_t)n[0];
    if ((size_t)n[1] > nmax) nmax = n[1];
    if ((size_t)n[2] > nmax) nmax = n[2];

    size_t off = 0;
    float* psum = (float*)((char*)d_ws + off); off += (size_t)3 * B_GRAPHS * 10 * sizeof(float);
    float* pcnt = (float*)((char*)d_ws + off); off += (size_t)3 * B_GRAPHS * sizeof(float);
    off = (off + 255) & ~(size_t)255;
    float* O = (float*)((char*)d_ws + off);    off += nmax * 96 * sizeof(float);
    off = (off + 255) & ~(size_t)255;
    float* Y = (float*)((char*)d_ws + off);    off += nmax * 32 * sizeof(float);
    (void)ws_size; (void)n_in; (void)out_size;

    hipMemsetAsync(psum, 0, (size_t)3 * B_GRAPHS * 11 * sizeof(float), stream); // psum+pcnt contiguous

    for (int t = 0; t < 3; ++t) {
        int nt = n[t], ne = nnz[t];

        // Stage 1: y1 = L @ X (F=3), o1 = leaky(y1 @ W1^T + b1) -> O[:,0:32]
        hipMemsetAsync(Y, 0, (size_t)nt * 3 * sizeof(float), stream);
        spmm_f3<<<(ne + 255) / 256, 256, 0, stream>>>(Lr[t], Lc[t], Lv[t], X[t], Y, ne);
        lin3_leaky<<<((unsigned)nt * 32 + 255) / 256, 256, 0, stream>>>(Y, CW1[t], Cb1[t], O, nt);

        // Stage 2: y2 = L @ o1, o2 = leaky(y2 @ W2^T + b2) -> O[:,32:64]
        hipMemsetAsync(Y, 0, (size_t)nt * 32 * sizeof(float), stream);
        spmm_f32<<<((unsigned)ne * 8u + 255u) / 256u, 256, 0, stream>>>(
            Lr[t], Lc[t], Lv[t], O + 0, 96, Y, ne);
        int waves = (nt + 15) / 16;
        gemm32_wmma_leaky<<<(waves + 7) / 8, 256, 0, stream>>>(Y, CW2[t], Cb2[t], O + 32, 96, nt);

        // Stage 3: y3 = L @ o2, o3 = leaky(y3 @ W3^T + b3) -> O[:,64:96]
        hipMemsetAsync(Y, 0, (size_t)nt * 32 * sizeof(float), stream);
        spmm_f32<<<((unsigned)ne * 8u + 255u) / 256u, 256, 0, stream>>>(
            Lr[t], Lc[t], Lv[t], O + 32, 96, Y, ne);
        gemm32_wmma_leaky<<<(waves + 7) / 8, 256, 0, stream>>>(Y, CW3[t], Cb3[t], O + 64, 96, nt);

        // Per-node 96->10 linear + mean-pool accumulation
        layer_pool<<<(nt + 255) / 256, 256, 0, stream>>>(
            O, layW[t], layb[t], bat[t],
            psum + (size_t)t * B_GRAPHS * 10, pcnt + (size_t)t * B_GRAPHS, nt);
    }

    final_combine<<<(B_GRAPHS + 255) / 256, 256, 0, stream>>>(psum, pcnt, combW, combb, (float*)d_out);
}